// GCN_18348100288854
// MI455X (gfx1250) — compile-verified
//
#include <hip/hip_runtime.h>

typedef __attribute__((ext_vector_type(2))) float v2f;
typedef __attribute__((ext_vector_type(8))) float v8f;

#define N_NODES 8192

// ---------------------------------------------------------------------------
// Small dense linear: out[n,o] = (relu?)( sum_k x[n,k]*W[k,o] + b[o] )
// Feature dims are tiny (<=128) -> scalar FMA loop; x rows are shared by
// `dout` consecutive threads so reads are cache-friendly. Negligible cost.
// ---------------------------------------------------------------------------
__global__ void gcn_lin_kernel(const float* __restrict__ x,
                               const float* __restrict__ W,
                               const float* __restrict__ b,
                               float* __restrict__ out,
                               int din, int dout, int do_relu) {
  int idx = blockIdx.x * blockDim.x + threadIdx.x;
  int n = idx / dout;
  int o = idx - n * dout;
  if (n >= N_NODES) return;
  const float* xr = x + (size_t)n * din;
  float acc = b[o];
  for (int k = 0; k < din; ++k)
    acc = fmaf(xr[k], W[k * dout + o], acc);
  if (do_relu) acc = fmaxf(acc, 0.0f);
  out[(size_t)n * dout + o] = acc;
}

// ---------------------------------------------------------------------------
// Aggregation: out = relu(adj @ h), adj [N,N] f32 (streams from HBM once per
// layer -> memory-bound term), h [N, D], D = 16*DT (reused by every wave ->
// staged in LDS per block).
//
// Per block (4 waves, 64 output rows): loop over K-tiles of KT=128.
//   stage:   h[k0..k0+KT) is a contiguous KT*D-float chunk -> coalesced
//            float4 global loads -> padded LDS (row stride D+4 dwords; keeps
//            16B alignment for b128 ds_stores, staggers banks for reads).
//   compute: per wave, per k-step of 4:
//            A (adj 16x4): lane m: v0=A[m][0],v1=A[m][1]; lane 16+m:
//              v0=A[m][2],v1=A[m][3]  -> one global_load_b64 per lane.
//            B (h 4x16) from LDS: lane m: rows k,k+1; lane 16+m: rows k+2,k+3
//            D accumulates in v8f tiles; all DT column-tiles per wave so adj
//            is read exactly once per layer.
//   C/D (16x16): lane (h=lane>>4, m=lane&15): VGPR j = D[j + 8*h][m]
// ---------------------------------------------------------------------------
template <int DT>
__global__ void gcn_agg_kernel(const float* __restrict__ adj,
                               const float* __restrict__ h,
                               float* __restrict__ out) {
  constexpr int D  = 16 * DT;
  constexpr int KT = 128;        // K-tile rows staged per block
  constexpr int DP = D + 4;      // padded LDS row stride (dwords)
  __shared__ float ldsB[KT * DP];

  const int tid  = threadIdx.x;  // 0..127, 4 waves
  const int lane = tid & 31;
  const int wave = tid >> 5;
  const int row0 = (blockIdx.x * 4 + wave) * 16;
  const int half = lane >> 4;    // K-pair select (A) / K-row pair select (B)
  const int m    = lane & 15;    // row (A) / column (B, C, D)

  v8f acc[DT];
#pragma unroll
  for (int t = 0; t < DT; ++t)
#pragma unroll
    for (int j = 0; j < 8; ++j) acc[t][j] = 0.0f;

  const float* arow = adj + (size_t)(row0 + m) * N_NODES + 2 * half;

  for (int k0 = 0; k0 < N_NODES; k0 += KT) {
    __syncthreads();  // previous tile's compute done before overwrite
    // ---- stage h[k0 .. k0+KT) into LDS (coalesced b128 loads) ----
    const float4* src = (const float4*)(h + (size_t)k0 * D);
#pragma unroll
    for (int r = 0; r < D / 4; ++r) {          // KT*D/4 float4s / 128 threads
      int i = tid + r * 128;
      float4 v = src[i];
      int kk = (4 * i) / D;
      int c  = (4 * i) % D;                    // multiple of 4 -> 16B aligned
      *(float4*)(&ldsB[kk * DP + c]) = v;
    }
    __syncthreads();

    // ---- compute on the tile ----
#pragma unroll 4
    for (int kk = 0; kk < KT; kk += 4) {
      v2f a = *(const v2f*)(arow + k0 + kk);   // global_load_b64 (stream)
      const float* hb = &ldsB[(kk + 2 * half) * DP + m];
#pragma unroll
      for (int t = 0; t < DT; ++t) {
        v2f b;
        b.x = hb[16 * t];                      // ds_load_b32
        b.y = hb[DP + 16 * t];                 // ds_load_b32 (next K row)
        acc[t] = __builtin_amdgcn_wmma_f32_16x16x4_f32(
            /*neg_a=*/false, a, /*neg_b=*/false, b,
            /*c_mod=*/(short)0, acc[t], /*reuse_a=*/false, /*reuse_b=*/false);
      }
    }
  }

#pragma unroll
  for (int t = 0; t < DT; ++t) {
#pragma unroll
    for (int j = 0; j < 8; ++j) {
      int r = row0 + j + 8 * half;
      out[(size_t)r * D + 16 * t + m] = fmaxf(acc[t][j], 0.0f);
    }
  }
}

// ---------------------------------------------------------------------------
// Pipeline:
//   wsA = x @ W0 + b0            [8192,16]
//   wsB = relu(adj @ wsA)        [8192,16]
//   wsA = wsB @ W1 + b1          [8192,32]
//   wsB = relu(adj @ wsA)        [8192,32]
//   wsA = wsB @ W2 + b2          [8192,64]
//   wsB = relu(adj @ wsA)        [8192,64]
//   wsA = relu(wsB @ Wo0 + bo0)  [8192,32]
//   out = wsA @ Wo1 + bo1        [8192,10]
// ---------------------------------------------------------------------------
extern "C" void kernel_launch(void* const* d_in, const int* in_sizes, int n_in,
                              void* d_out, int out_size, void* d_ws, size_t ws_size,
                              hipStream_t stream) {
  const float* x   = (const float*)d_in[0];
  const float* adj = (const float*)d_in[1];
  const float* W0  = (const float*)d_in[2];
  const float* b0  = (const float*)d_in[3];
  const float* W1  = (const float*)d_in[4];
  const float* b1  = (const float*)d_in[5];
  const float* W2  = (const float*)d_in[6];
  const float* b2  = (const float*)d_in[7];
  const float* Wo0 = (const float*)d_in[8];
  const float* bo0 = (const float*)d_in[9];
  const float* Wo1 = (const float*)d_in[10];
  const float* bo1 = (const float*)d_in[11];
  float* out = (float*)d_out;

  float* wsA = (float*)d_ws;                 // 8192*64 floats = 2 MB
  float* wsB = wsA + (size_t)N_NODES * 64;   // 8192*64 floats = 2 MB

  const int LIN_BLK  = 256;
  const int AGG_BLK  = 128;                  // 4 waves/block, 64 rows/block
  const int AGG_GRID = N_NODES / 64;         // 128 blocks

  // Layer 1: 128 -> 16
  gcn_lin_kernel<<<(N_NODES * 16 + LIN_BLK - 1) / LIN_BLK, LIN_BLK, 0, stream>>>(
      x, W0, b0, wsA, 128, 16, 0);
  gcn_agg_kernel<1><<<AGG_GRID, AGG_BLK, 0, stream>>>(adj, wsA, wsB);

  // Layer 2: 16 -> 32
  gcn_lin_kernel<<<(N_NODES * 32 + LIN_BLK - 1) / LIN_BLK, LIN_BLK, 0, stream>>>(
      wsB, W1, b1, wsA, 16, 32, 0);
  gcn_agg_kernel<2><<<AGG_GRID, AGG_BLK, 0, stream>>>(adj, wsA, wsB);

  // Layer 3: 32 -> 64
  gcn_lin_kernel<<<(N_NODES * 64 + LIN_BLK - 1) / LIN_BLK, LIN_BLK, 0, stream>>>(
      wsB, W2, b2, wsA, 32, 64, 0);
  gcn_agg_kernel<4><<<AGG_GRID, AGG_BLK, 0, stream>>>(adj, wsA, wsB);

  // Head: 64 -> 32 (relu) -> 10
  gcn_lin_kernel<<<(N_NODES * 32 + LIN_BLK - 1) / LIN_BLK, LIN_BLK, 0, stream>>>(
      wsB, Wo0, bo0, wsA, 64, 32, 1);
  gcn_lin_kernel<<<(N_NODES * 10 + LIN_BLK - 1) / LIN_BLK, LIN_BLK, 0, stream>>>(
      wsA, Wo1, bo1, out, 32, 10, 0);
}